// PPISPSlangtorch_36515811950648
// MI455X (gfx1250) — compile-verified
//
#include <hip/hip_runtime.h>
#include <hip/hip_bf16.h>
#include <stdint.h>

// ---------------------------------------------------------------------------
// gfx1250 async global->LDS copy helpers (inline asm; ASYNCcnt-tracked DMA).
// VDST operand carries the LDS byte address; hardware adds LDS_BASE.
// Generic pointers to __shared__ carry the LDS offset in addr[31:0] (ISA 10.2),
// so truncating the generic address yields the LDS byte offset.
// ---------------------------------------------------------------------------
__device__ __forceinline__ void async_lds_b128(void* lds_dst, const void* gsrc) {
    uint32_t lds_off = (uint32_t)(uintptr_t)lds_dst;
    asm volatile("global_load_async_to_lds_b128 %0, %1, off"
                 :: "v"(lds_off), "v"(gsrc) : "memory");
}
__device__ __forceinline__ void async_lds_b32(void* lds_dst, const void* gsrc) {
    uint32_t lds_off = (uint32_t)(uintptr_t)lds_dst;
    asm volatile("global_load_async_to_lds_b32 %0, %1, off"
                 :: "v"(lds_off), "v"(gsrc) : "memory");
}
__device__ __forceinline__ void wait_async_zero() {
    asm volatile("s_wait_asynccnt 0" ::: "memory");
}

__device__ __forceinline__ void stage_table(float* dst, const float* src, int n,
                                            int tid, int nthr) {
    const int nv = n >> 2;                           // 16B chunks
    for (int i = tid; i < nv; i += nthr)
        async_lds_b128(dst + 4 * i, src + 4 * i);
    for (int i = (nv << 2) + tid; i < n; i += nthr)  // scalar tail
        async_lds_b32(dst + i, src + i);
}

// Numerically stable softplus, matching jax.nn.softplus.
__device__ __forceinline__ float softplus_f(float x) {
    return fmaxf(x, 0.0f) + log1pf(expf(-fabsf(x)));
}

__device__ __forceinline__ int pad4i(int x) { return (x + 3) & ~3; }

// Full per-point ISP pipeline, reading all parameters from LDS tables.
__device__ __forceinline__ void process_point(
    float r, float g, float b, float u, float v, int f, int cam,
    const float* __restrict__ s_exp, const float* __restrict__ s_col,
    const float* __restrict__ s_vig, const float* __restrict__ s_crf,
    float* __restrict__ o /* [3] */)
{
    // exposure
    const float e = exp2f(s_exp[f]);
    float x0 = r * e, x1 = g * e, x2 = b * e;

    // vignetting (per-channel radial polynomial)
    const float* vp = s_vig + cam * 15;
    {
        float du = u - vp[0], dv = v - vp[1];
        float r2 = du * du + dv * dv;
        x0 *= 1.0f + r2 * (vp[2] + r2 * (vp[3] + r2 * vp[4]));
        du = u - vp[5]; dv = v - vp[6];
        r2 = du * du + dv * dv;
        x1 *= 1.0f + r2 * (vp[7] + r2 * (vp[8] + r2 * vp[9]));
        du = u - vp[10]; dv = v - vp[11];
        r2 = du * du + dv * dv;
        x2 *= 1.0f + r2 * (vp[12] + r2 * (vp[13] + r2 * vp[14]));
    }

    // per-frame color: white balance + 3x3 mixing matrix
    const float* cp = s_col + f * 8;
    x0 *= expf(cp[0]);
    x2 *= expf(cp[1]);
    const float o0 = cp[2], o1 = cp[3], o2 = cp[4],
                o3 = cp[5], o4 = cp[6], o5 = cp[7];
    float y[3];
    y[0] = (1.0f - o0 - o1) * x0 + o0 * x1 + o1 * x2;
    y[1] = o2 * x0 + (1.0f - o2 - o3) * x1 + o3 * x2;
    y[2] = o4 * x0 + o5 * x1 + (1.0f - o4 - o5) * x2;

    // per-camera CRF
    const float* kp = s_crf + cam * 12;
#pragma unroll
    for (int ch = 0; ch < 3; ++ch) {
        const float* k = kp + ch * 4;
        const float a  = softplus_f(k[0]) + 0.3f;
        const float bb = softplus_f(k[1]) + 0.3f;
        const float cc = softplus_f(k[2]) + 0.1f;
        const float d  = k[3];
        const float xp = powf(fmaxf(y[ch], 1e-6f), a);
        const float s  = 1.0f / (1.0f + expf(-(xp - d) / cc));
        o[ch] = bb * s;
    }
}

extern "C" __global__ void __launch_bounds__(256)
ppisp_kernel(const float* __restrict__ expo,    // [F]
             const float* __restrict__ vig,     // [C*3*5]
             const float* __restrict__ col,     // [F*8]
             const float* __restrict__ crf,     // [C*3*4]
             const float* __restrict__ rgb,     // [B*3]
             const float* __restrict__ pix,     // [B*2]
             const int*   __restrict__ cams,    // [B]
             const int*   __restrict__ frames,  // [B]
             const int*   __restrict__ wp,      // [1]
             const int*   __restrict__ hp,      // [1]
             float*       __restrict__ out,     // [B*3]
             int B, int F, int C)
{
    extern __shared__ float smem[];
    float* s_exp = smem;
    float* s_col = s_exp + pad4i(F);
    float* s_vig = s_col + pad4i(8 * F);
    float* s_crf = s_vig + pad4i(15 * C);

    const int tid  = threadIdx.x;
    const int nthr = blockDim.x;

    // ---- stage all parameter tables into LDS via the async DMA path ----
    stage_table(s_exp, expo, F,      tid, nthr);
    stage_table(s_col, col,  8 * F,  tid, nthr);
    stage_table(s_vig, vig,  15 * C, tid, nthr);
    stage_table(s_crf, crf,  12 * C, tid, nthr);
    wait_async_zero();     // this wave's async copies landed in LDS
    __syncthreads();       // every wave's copies visible to the workgroup

    const float sx = 2.0f / (float)(*wp);
    const float sy = 2.0f / (float)(*hp);

    const int gtid   = blockIdx.x * blockDim.x + tid;
    const int stride = gridDim.x * blockDim.x;

    // ---- vectorized main loop: 4 points / thread, all b128 traffic ----
    const float4* __restrict__ rgb4 = (const float4*)rgb;
    const float4* __restrict__ pix4 = (const float4*)pix;
    const int4*   __restrict__ cam4 = (const int4*)cams;
    const int4*   __restrict__ frm4 = (const int4*)frames;
    float4*       __restrict__ out4 = (float4*)out;

    const int nchunks = B >> 2;
    for (int c = gtid; c < nchunks; c += stride) {
        const float4 r0 = rgb4[3 * c + 0];
        const float4 r1 = rgb4[3 * c + 1];
        const float4 r2 = rgb4[3 * c + 2];
        const float4 p0 = pix4[2 * c + 0];
        const float4 p1 = pix4[2 * c + 1];
        const int4   cm = cam4[c];
        const int4   fr = frm4[c];

        const float rr[4][3] = {{r0.x, r0.y, r0.z}, {r0.w, r1.x, r1.y},
                                {r1.z, r1.w, r2.x}, {r2.y, r2.z, r2.w}};
        const float uu[4] = {p0.x * sx - 1.0f, p0.z * sx - 1.0f,
                             p1.x * sx - 1.0f, p1.z * sx - 1.0f};
        const float vv[4] = {p0.y * sy - 1.0f, p0.w * sy - 1.0f,
                             p1.y * sy - 1.0f, p1.w * sy - 1.0f};
        const int cmi[4] = {cm.x, cm.y, cm.z, cm.w};
        const int fri[4] = {fr.x, fr.y, fr.z, fr.w};

        float o[12];
#pragma unroll
        for (int p = 0; p < 4; ++p)
            process_point(rr[p][0], rr[p][1], rr[p][2], uu[p], vv[p],
                          fri[p], cmi[p], s_exp, s_col, s_vig, s_crf, o + 3 * p);

        out4[3 * c + 0] = make_float4(o[0], o[1], o[2],  o[3]);
        out4[3 * c + 1] = make_float4(o[4], o[5], o[6],  o[7]);
        out4[3 * c + 2] = make_float4(o[8], o[9], o[10], o[11]);
    }

    // ---- scalar tail for B % 4 points ----
    for (int i = (nchunks << 2) + gtid; i < B; i += stride) {
        const float* rp = rgb + 3 * i;
        const float2 pxy = *(const float2*)(pix + 2 * i);
        float o[3];
        process_point(rp[0], rp[1], rp[2],
                      pxy.x * sx - 1.0f, pxy.y * sy - 1.0f,
                      frames[i], cams[i], s_exp, s_col, s_vig, s_crf, o);
        float* op = out + 3 * i;
        op[0] = o[0]; op[1] = o[1]; op[2] = o[2];
    }
}

extern "C" void kernel_launch(void* const* d_in, const int* in_sizes, int n_in,
                              void* d_out, int out_size, void* d_ws, size_t ws_size,
                              hipStream_t stream) {
    (void)n_in; (void)d_ws; (void)ws_size; (void)out_size;

    const float* expo   = (const float*)d_in[0];
    const float* vig    = (const float*)d_in[1];
    const float* col    = (const float*)d_in[2];
    const float* crf    = (const float*)d_in[3];
    const float* rgb    = (const float*)d_in[4];
    const float* pix    = (const float*)d_in[5];
    const int*   cams   = (const int*)d_in[6];
    const int*   frames = (const int*)d_in[7];
    const int*   wp     = (const int*)d_in[8];
    const int*   hp     = (const int*)d_in[9];
    float*       out    = (float*)d_out;

    const int F = in_sizes[0];          // exposure_params: [F]
    const int C = in_sizes[1] / 15;     // vignetting: [C,3,5]
    const int B = in_sizes[4] / 3;      // rgb: [B,3]

    auto pad4 = [](int x) { return (x + 3) & ~3; };
    const size_t smem_bytes =
        (size_t)(pad4(F) + pad4(8 * F) + pad4(15 * C) + pad4(12 * C)) * sizeof(float);

    const int threads = 256;            // 8 wave32 per block
    const int nchunks = B / 4;
    int blocks = (nchunks + threads - 1) / threads;
    if (blocks > 2048) blocks = 2048;   // persistent grid: stage LDS once per block
    if (blocks < 1) blocks = 1;

    ppisp_kernel<<<blocks, threads, smem_bytes, stream>>>(
        expo, vig, col, crf, rgb, pix, cams, frames, wp, hp, out, B, F, C);
}